// SAGEModel_50079318671418
// MI455X (gfx1250) — compile-verified
//
#include <hip/hip_runtime.h>

#define N_NODES 100000
#define N_EDGES 625000
#define HIDDEN  128
#define D_OUT_N 64
#define LN_EPS  1e-5f

typedef __attribute__((ext_vector_type(2))) float v2f;
typedef __attribute__((ext_vector_type(4))) float v4f;
typedef __attribute__((ext_vector_type(8))) float v8f;

__global__ void zero_kernel(float* __restrict__ p, int n) {
    int i = blockIdx.x * blockDim.x + threadIdx.x;
    if (i < n) p[i] = 0.0f;
}

__global__ void degree_kernel(const int* __restrict__ dst, float* __restrict__ deg) {
    int e = blockIdx.x * blockDim.x + threadIdx.x;
    if (e < N_EDGES) atomicAdd(&deg[dst[e]], 1.0f);
}

// one wave per edge: gather x[src] row, atomic-accumulate into agg[dst] row
__global__ void scatter_kernel(const float* __restrict__ x, const int* __restrict__ src,
                               const int* __restrict__ dst, float* __restrict__ agg) {
    int wave = blockIdx.x * (blockDim.x >> 5) + (threadIdx.x >> 5);
    int lane = threadIdx.x & 31;
    if (wave >= N_EDGES) return;
    int s = src[wave];
    int d = dst[wave];
    const float* xs = x + (size_t)s * HIDDEN;
    float* ad = agg + (size_t)d * HIDDEN;
#pragma unroll
    for (int i = 0; i < 4; ++i) {
        int c = i * 32 + lane;
        atomicAdd(&ad[c], xs[c]);
    }
}

// Fused: h = (agg*invdeg)@Wl^T + bl + x@Wr^T (+x) ; LayerNorm ; ReLU
// block = 256 threads = 8 waves; 16 nodes per block; wave w owns cols [16w,16w+16)
__launch_bounds__(256)
__global__ void sage_layer_kernel(const float* __restrict__ xin,
                                  const float* __restrict__ agg,
                                  const float* __restrict__ deg,
                                  const float* __restrict__ Wl,
                                  const float* __restrict__ bl,
                                  const float* __restrict__ Wr,
                                  const float* __restrict__ gamma,
                                  const float* __restrict__ beta,
                                  float* __restrict__ xout,
                                  int residual) {
    __shared__ __align__(16) float s_agg[16 * HIDDEN];
    __shared__ __align__(16) float s_x  [16 * HIDDEN];
    __shared__ __align__(16) float s_h  [16 * HIDDEN];

    const int tid = threadIdx.x;
    const size_t row0 = (size_t)blockIdx.x * 16;

    // stage 16x128 agg (scaled by 1/max(deg,1)) and x tiles: 512 float4 loads
    const v4f* agg4 = (const v4f*)(agg + row0 * HIDDEN);
    const v4f* x4   = (const v4f*)(xin + row0 * HIDDEN);
    v4f* sagg4 = (v4f*)s_agg;
    v4f* sx4   = (v4f*)s_x;
#pragma unroll
    for (int i = 0; i < 2; ++i) {
        int f = tid + i * 256;          // float4 index in [0,512)
        int r = f >> 5;                 // row of the 16
        float invd = 1.0f / fmaxf(deg[row0 + r], 1.0f);
        v4f a = agg4[f];
        a.x *= invd; a.y *= invd; a.z *= invd; a.w *= invd;
        sagg4[f] = a;
        sx4[f]   = x4[f];
    }
    __syncthreads();

    const int wave  = tid >> 5;
    const int lane  = tid & 31;
    const int half  = lane >> 4;        // 0: K+0/K+1 (A) rows M   | 1: K+2/K+3, M+8 for C/D
    const int mn    = lane & 15;        // M for A-frag, N for B/C/D-frag
    const int jbase = wave * 16;

    const float* wlrow = Wl + (size_t)(jbase + mn) * HIDDEN;  // B[k][n] = Wl[n][k]
    const float* wrrow = Wr + (size_t)(jbase + mn) * HIDDEN;

    v8f acc = {};
#pragma unroll
    for (int kk = 0; kk < HIDDEN; kk += 4) {
        int ko = kk + 2 * half;
        v2f a_agg = *(const v2f*)&s_agg[mn * HIDDEN + ko];
        v2f a_x   = *(const v2f*)&s_x  [mn * HIDDEN + ko];
        v2f b_l   = *(const v2f*)&wlrow[ko];
        v2f b_r   = *(const v2f*)&wrrow[ko];
        acc = __builtin_amdgcn_wmma_f32_16x16x4_f32(false, a_agg, false, b_l,
                                                    (short)0, acc, false, false);
        acc = __builtin_amdgcn_wmma_f32_16x16x4_f32(false, a_x,   false, b_r,
                                                    (short)0, acc, false, false);
    }

    // write D tile (+bias, +residual) into LDS: lane holds (M = v+8*half, N = mn)
    float bias = bl[jbase + mn];
#pragma unroll
    for (int v = 0; v < 8; ++v) {
        int m = v + 8 * half;
        float h = acc[v] + bias;
        if (residual) h += s_x[m * HIDDEN + jbase + mn];
        s_h[m * HIDDEN + jbase + mn] = h;
    }
    __syncthreads();

    // LayerNorm + ReLU: 16 threads per row, 8 contiguous cols each; wave32 shfl width 16
    int r  = tid >> 4;
    int cb = (tid & 15) * 8;
    float sum = 0.0f, sumsq = 0.0f;
    float hv[8];
#pragma unroll
    for (int i = 0; i < 8; ++i) {
        float h = s_h[r * HIDDEN + cb + i];
        hv[i] = h;
        sum += h;
        sumsq += h * h;
    }
#pragma unroll
    for (int off = 8; off >= 1; off >>= 1) {
        sum   += __shfl_xor(sum,   off, 16);
        sumsq += __shfl_xor(sumsq, off, 16);
    }
    float mu   = sum * (1.0f / HIDDEN);
    float var  = sumsq * (1.0f / HIDDEN) - mu * mu;
    float rstd = rsqrtf(var + LN_EPS);
    float* orow = xout + (row0 + r) * HIDDEN;
#pragma unroll
    for (int i = 0; i < 8; ++i) {
        int c = cb + i;
        float v = (hv[i] - mu) * rstd * gamma[c] + beta[c];
        orow[c] = fmaxf(v, 0.0f);
    }
}

// out = x @ Wo^T + bo ; block = 128 threads = 4 waves, 16 nodes, wave w -> cols [16w,16w+16)
__launch_bounds__(128)
__global__ void out_proj_kernel(const float* __restrict__ xin,
                                const float* __restrict__ Wo,
                                const float* __restrict__ bo,
                                float* __restrict__ out) {
    __shared__ __align__(16) float s_x[16 * HIDDEN];
    const int tid = threadIdx.x;
    const size_t row0 = (size_t)blockIdx.x * 16;

    const v4f* x4 = (const v4f*)(xin + row0 * HIDDEN);
    v4f* sx4 = (v4f*)s_x;
#pragma unroll
    for (int i = 0; i < 4; ++i) sx4[tid + i * 128] = x4[tid + i * 128];
    __syncthreads();

    const int wave  = tid >> 5;
    const int lane  = tid & 31;
    const int half  = lane >> 4;
    const int mn    = lane & 15;
    const int jbase = wave * 16;

    const float* worow = Wo + (size_t)(jbase + mn) * HIDDEN;
    v8f acc = {};
#pragma unroll
    for (int kk = 0; kk < HIDDEN; kk += 4) {
        int ko = kk + 2 * half;
        v2f a = *(const v2f*)&s_x[mn * HIDDEN + ko];
        v2f b = *(const v2f*)&worow[ko];
        acc = __builtin_amdgcn_wmma_f32_16x16x4_f32(false, a, false, b,
                                                    (short)0, acc, false, false);
    }
    float bias = bo[jbase + mn];
#pragma unroll
    for (int v = 0; v < 8; ++v) {
        int m = v + 8 * half;
        out[(row0 + m) * D_OUT_N + jbase + mn] = acc[v] + bias;
    }
}

extern "C" void kernel_launch(void* const* d_in, const int* in_sizes, int n_in,
                              void* d_out, int out_size, void* d_ws, size_t ws_size,
                              hipStream_t stream) {
    const float* x     = (const float*)d_in[0];
    const int*   ei    = (const int*)  d_in[1];
    const float* Wl    = (const float*)d_in[2];
    const float* bl    = (const float*)d_in[3];
    const float* Wr    = (const float*)d_in[4];
    const float* gamma = (const float*)d_in[5];
    const float* beta  = (const float*)d_in[6];
    const float* Wo    = (const float*)d_in[7];
    const float* bo    = (const float*)d_in[8];
    float* out = (float*)d_out;

    const int* src = ei;
    const int* dst = ei + N_EDGES;

    // workspace layout (floats): deg[N] | agg[N*H] | buf0[N*H] | buf1[N*H]
    float* ws   = (float*)d_ws;
    float* deg  = ws;
    float* aggb = ws + 102400;                       // 16B-aligned pad past N_NODES
    float* buf0 = aggb + (size_t)N_NODES * HIDDEN;
    float* buf1 = buf0 + (size_t)N_NODES * HIDDEN;

    const int NH = N_NODES * HIDDEN;

    zero_kernel<<<(N_NODES + 255) / 256, 256, 0, stream>>>(deg, N_NODES);
    degree_kernel<<<(N_EDGES + 255) / 256, 256, 0, stream>>>(dst, deg);

    const float* cur = x;
    float* nxt = buf0;
    for (int layer = 0; layer < 3; ++layer) {
        zero_kernel<<<(NH + 255) / 256, 256, 0, stream>>>(aggb, NH);
        scatter_kernel<<<(N_EDGES + 7) / 8, 256, 0, stream>>>(cur, src, dst, aggb);
        sage_layer_kernel<<<N_NODES / 16, 256, 0, stream>>>(
            cur, aggb, deg,
            Wl + (size_t)layer * HIDDEN * HIDDEN, bl + layer * HIDDEN,
            Wr + (size_t)layer * HIDDEN * HIDDEN,
            gamma + layer * HIDDEN, beta + layer * HIDDEN,
            nxt, layer > 0 ? 1 : 0);
        cur = nxt;
        nxt = (cur == buf0) ? buf1 : buf0;
    }
    out_proj_kernel<<<N_NODES / 16, 128, 0, stream>>>(cur, Wo, bo, out);
}